// FullModelWithMoments_40853728919713
// MI455X (gfx1250) — compile-verified
//
#include <hip/hip_runtime.h>
#include <math.h>

#define NN 1024
#define BB 1024
#define PIf 3.14159265358979323846f
#define KC 64          // K-chunk staged in LDS
#define BSTR 40        // LDS row stride in floats (16B aligned rows, bank-safe)

typedef float v2f __attribute__((ext_vector_type(2)));
typedef float v4f __attribute__((ext_vector_type(4)));
typedef float v8f __attribute__((ext_vector_type(8)));
typedef int   v4i __attribute__((ext_vector_type(4)));

__device__ __forceinline__ v8f wmma_f32(v2f a, v2f b, v8f c) {
  // V_WMMA_F32_16X16X4_F32 : D = A(16x4) * B(4x16) + C(16x16), full f32
  return __builtin_amdgcn_wmma_f32_16x16x4_f32(false, a, false, b, (short)0, c, false, false);
}

// --------------------------------------------------------------------------
// CDNA5 async global->LDS copy (ASYNCcnt) with compile-safe fallback.
// Builtin signature (from hipcc diagnostic): (v4i AS1*, v4i AS3*, imm, imm)
// --------------------------------------------------------------------------
#if defined(__AMDGCN__) && __has_builtin(__builtin_amdgcn_global_load_async_to_lds_b128)
#define ASYNC_LDS_COPY 1
#else
#define ASYNC_LDS_COPY 0
#endif

#if ASYNC_LDS_COPY
typedef __attribute__((address_space(1))) v4i* gbl_v4i_p;
typedef __attribute__((address_space(3))) v4i* lds_v4i_p;
#endif

__device__ __forceinline__ void copy16_to_lds(const float* g, float* l) {
#if ASYNC_LDS_COPY
  // generic LDS pointer: low 32 bits are the LDS byte offset (ISA 10.2)
  __builtin_amdgcn_global_load_async_to_lds_b128(
      (gbl_v4i_p)(size_t)g, (lds_v4i_p)(unsigned)(size_t)l, 0, 0);
#else
  *(v4f*)l = *(const v4f*)g;
#endif
}

__device__ __forceinline__ void async_join() {
#if ASYNC_LDS_COPY
#if __has_builtin(__builtin_amdgcn_s_wait_asynccnt)
  __builtin_amdgcn_s_wait_asynccnt(0);
#else
  asm volatile("s_wait_asynccnt 0" ::: "memory");
#endif
#endif
}

// Stage a KC x 32 chunk of both cos and sin matrices into LDS.
// Chunk element (r,c) = M[(nb+r)*NN + k0 + c]; LDS layout r*BSTR + c.
__device__ __forceinline__ void stage_chunk(
    const float* __restrict__ cm, const float* __restrict__ sm,
    int nb, int k0, float* sC, float* sS, int t) {
  for (int q = t; q < 512; q += 256) {           // 512 16B quads per matrix
    const int r = q >> 3, c4 = (q & 7) * 4;
    const size_t go = (size_t)(nb + r) * NN + k0 + c4;
    const int lo = r * BSTR + c4;
    copy16_to_lds(cm + go, sC + lo);
    copy16_to_lds(sm + go, sS + lo);
  }
}

// ---------------------------------------------------------------------------
// Stage 1: Xre[b,k] = (sum_n x[b,0,n] cos[n,k]) * h[k]
//          Xim[b,k] = -(sum_n x[b,0,n] sin[n,k]) * h[k]
// Block = 8 waves = 128(b) x 32(k) macro-tile; B staged in LDS, double-buffered.
// ---------------------------------------------------------------------------
__global__ void __launch_bounds__(256) k_dft_fwd(
    const float* __restrict__ x, const float* __restrict__ cm,
    const float* __restrict__ sm, const float* __restrict__ h,
    float* __restrict__ Xre, float* __restrict__ Xim) {
  __shared__ float sC[2][KC * BSTR];
  __shared__ float sS[2][KC * BSTR];
  const int t = threadIdx.x;
  const int wave = t >> 5, lane = t & 31;
  const int b0 = (blockIdx.x >> 5) * 128 + wave * 16;  // 8 row-blocks
  const int k0 = (blockIdx.x & 31) * 32;               // 32 col-blocks
  const int row = lane & 15, kh = lane >> 4, col = row;
  const float* xrow = x + (size_t)(b0 + row) * (2 * NN); // channel 0
  v8f aC0 = {}, aS0 = {}, aC1 = {}, aS1 = {};
  stage_chunk(cm, sm, 0, k0, sC[0], sS[0], t);
  for (int ch = 0; ch < NN / KC; ++ch) {
    const int cur = ch & 1;
    async_join();
    __syncthreads();
    if (ch + 1 < NN / KC)
      stage_chunk(cm, sm, (ch + 1) * KC, k0, sC[cur ^ 1], sS[cur ^ 1], t);
    const float* bc_ = sC[cur];
    const float* bs_ = sS[cur];
    const int nbase = ch * KC;
    for (int ks = 0; ks < KC / 4; ++ks) {
      const int R = ks * 4 + 2 * kh;
      v2f a = *(const v2f*)(xrow + nbase + R);
      v2f c0 = { bc_[R * BSTR + col],      bc_[(R + 1) * BSTR + col] };
      v2f s0 = { bs_[R * BSTR + col],      bs_[(R + 1) * BSTR + col] };
      v2f c1 = { bc_[R * BSTR + 16 + col], bc_[(R + 1) * BSTR + 16 + col] };
      v2f s1 = { bs_[R * BSTR + 16 + col], bs_[(R + 1) * BSTR + 16 + col] };
      aC0 = wmma_f32(a, c0, aC0);
      aS0 = wmma_f32(a, s0, aS0);
      aC1 = wmma_f32(a, c1, aC1);
      aS1 = wmma_f32(a, s1, aS1);
    }
    __syncthreads();
  }
  for (int ct = 0; ct < 2; ++ct) {
    const int kcol = k0 + ct * 16 + col;
    const float hv = h[kcol];
    const v8f accC = ct ? aC1 : aC0;
    const v8f accS = ct ? aS1 : aS0;
    for (int r = 0; r < 8; ++r) {
      const int b = b0 + r + 8 * kh;
      Xre[(size_t)b * NN + kcol] = accC[r] * hv;
      Xim[(size_t)b * NN + kcol] = -(accS[r] * hv);
    }
  }
}

// ---------------------------------------------------------------------------
// Stage 2: are = (Xre*cos - Xim*sin)/N ; aim = (Xre*sin + Xim*cos)/N
// ---------------------------------------------------------------------------
__global__ void __launch_bounds__(256) k_dft_inv(
    const float* __restrict__ Xre, const float* __restrict__ Xim,
    const float* __restrict__ cm, const float* __restrict__ sm,
    float* __restrict__ are, float* __restrict__ aim) {
  __shared__ float sC[2][KC * BSTR];
  __shared__ float sS[2][KC * BSTR];
  const int t = threadIdx.x;
  const int wave = t >> 5, lane = t & 31;
  const int b0 = (blockIdx.x >> 5) * 128 + wave * 16;
  const int n0 = (blockIdx.x & 31) * 32;
  const int row = lane & 15, kh = lane >> 4, col = row;
  const float* rrow = Xre + (size_t)(b0 + row) * NN;
  const float* irow = Xim + (size_t)(b0 + row) * NN;
  v8f rc0 = {}, rs0 = {}, ic0 = {}, is0 = {};
  v8f rc1 = {}, rs1 = {}, ic1 = {}, is1 = {};
  stage_chunk(cm, sm, 0, n0, sC[0], sS[0], t);
  for (int ch = 0; ch < NN / KC; ++ch) {
    const int cur = ch & 1;
    async_join();
    __syncthreads();
    if (ch + 1 < NN / KC)
      stage_chunk(cm, sm, (ch + 1) * KC, n0, sC[cur ^ 1], sS[cur ^ 1], t);
    const float* bc_ = sC[cur];
    const float* bs_ = sS[cur];
    const int kbase = ch * KC;
    for (int ks = 0; ks < KC / 4; ++ks) {
      const int R = ks * 4 + 2 * kh;
      v2f ar = *(const v2f*)(rrow + kbase + R);
      v2f ai = *(const v2f*)(irow + kbase + R);
      v2f c0 = { bc_[R * BSTR + col],      bc_[(R + 1) * BSTR + col] };
      v2f s0 = { bs_[R * BSTR + col],      bs_[(R + 1) * BSTR + col] };
      v2f c1 = { bc_[R * BSTR + 16 + col], bc_[(R + 1) * BSTR + 16 + col] };
      v2f s1 = { bs_[R * BSTR + 16 + col], bs_[(R + 1) * BSTR + 16 + col] };
      rc0 = wmma_f32(ar, c0, rc0);
      is0 = wmma_f32(ai, s0, is0);
      rs0 = wmma_f32(ar, s0, rs0);
      ic0 = wmma_f32(ai, c0, ic0);
      rc1 = wmma_f32(ar, c1, rc1);
      is1 = wmma_f32(ai, s1, is1);
      rs1 = wmma_f32(ar, s1, rs1);
      ic1 = wmma_f32(ai, c1, ic1);
    }
    __syncthreads();
  }
  const float invN = 1.0f / (float)NN;
  for (int ct = 0; ct < 2; ++ct) {
    const int ncol = n0 + ct * 16 + col;
    const v8f aRC = ct ? rc1 : rc0, aIS = ct ? is1 : is0;
    const v8f aRS = ct ? rs1 : rs0, aIC = ct ? ic1 : ic0;
    for (int r = 0; r < 8; ++r) {
      const int b = b0 + r + 8 * kh;
      are[(size_t)b * NN + ncol] = (aRC[r] - aIS[r]) * invN;
      aim[(size_t)b * NN + ncol] = (aRS[r] + aIC[r]) * invN;
    }
  }
}

// ---------------------------------------------------------------------------
// Block-wide reductions (wave32 shuffles + LDS broadcast)
// ---------------------------------------------------------------------------
__device__ __forceinline__ float bsum(float v, float* red) {
  #pragma unroll
  for (int off = 16; off > 0; off >>= 1) v += __shfl_xor(v, off, 32);
  const int t = threadIdx.x;
  __syncthreads();
  if ((t & 31) == 0) red[t >> 5] = v;
  __syncthreads();
  float s = 0.f;
  #pragma unroll
  for (int w = 0; w < 8; ++w) s += red[w];
  return s;
}
__device__ __forceinline__ float bmax(float v, float* red) {
  #pragma unroll
  for (int off = 16; off > 0; off >>= 1) v = fmaxf(v, __shfl_xor(v, off, 32));
  const int t = threadIdx.x;
  __syncthreads();
  if ((t & 31) == 0) red[t >> 5] = v;
  __syncthreads();
  float s = -INFINITY;
  #pragma unroll
  for (int w = 0; w < 8; ++w) s = fmaxf(s, red[w]);
  return s;
}

struct c2f { float r, i; };
__device__ __forceinline__ c2f cmul(c2f a, c2f b) { return { a.r*b.r - a.i*b.i, a.r*b.i + a.i*b.r }; }
__device__ __forceinline__ c2f cadd(c2f a, c2f b) { return { a.r + b.r, a.i + b.i }; }
__device__ __forceinline__ c2f cax(float s, c2f a) { return { s * a.r, s * a.i }; }

// ---------------------------------------------------------------------------
// Stage 3: per batch: amp/phase/unwrap/freq, central moments, cumulants,
// mom(32) -> clip -> W_mom matmul -> bn -> relu, write into zbuf[:,128:160]
// ---------------------------------------------------------------------------
__global__ void __launch_bounds__(256) k_moments(
    const float* __restrict__ x, const float* __restrict__ are,
    const float* __restrict__ aim, const float* __restrict__ Wm,
    const float* __restrict__ gm, const float* __restrict__ bm,
    float* __restrict__ zbuf) {
  __shared__ float sP[NN];
  __shared__ float sA[NN];
  __shared__ float sU[NN];
  __shared__ float part[256];
  __shared__ float red[8];
  __shared__ float smom[32];
  const int b = blockIdx.x, t = threadIdx.x;
  const float* ar = are + (size_t)b * NN;
  const float* ai = aim + (size_t)b * NN;

  for (int j = t; j < NN; j += 256) {
    const float r = ar[j], i = ai[j];
    sA[j] = sqrtf(r * r + i * i);
    const float sx = (float)(r > 0.f) - (float)(r < 0.f);
    const float sy = (float)(i > 0.f) - (float)(i < 0.f);
    sP[j] = sx * sx * atanf(i / r) + (1.f - sx) * 0.5f * (1.f + sy - sy * sy) * PIf;
  }
  __syncthreads();
  for (int j = t; j < NN; j += 256) {
    const float dx = (j == 0) ? 0.f : sP[j] - sP[j - 1];
    float md = fmodf(dx + PIf, 2.f * PIf);
    if (md < 0.f) md += 2.f * PIf;
    float dxm = md - PIf;
    if (dxm == -PIf && dx > 0.f) dxm = PIf;
    sU[j] = (fabsf(dx) < PIf) ? 0.f : (dxm - dx);
  }
  __syncthreads();
  {
    const int base = t * 4;
    float s = 0.f;
    for (int q = 0; q < 4; ++q) s += sU[base + q];
    part[t] = s;
    __syncthreads();
    for (int off = 1; off < 256; off <<= 1) {
      const float v = part[t];
      const float add = (t >= off) ? part[t - off] : 0.f;
      __syncthreads();
      part[t] = v + add;
      __syncthreads();
    }
    float c = (t > 0) ? part[t - 1] : 0.f;
    float tmp[4];
    for (int q = 0; q < 4; ++q) { c += sU[base + q]; tmp[q] = c; }
    for (int q = 0; q < 4; ++q) sU[base + q] = sP[base + q] + tmp[q];
  }
  __syncthreads();

  float s_amp = 0, s_uph = 0, s_frq = 0, s_a2 = 0, amx = -INFINITY;
  float M[22];
  #pragma unroll
  for (int k = 0; k < 22; ++k) M[k] = 0.f;
  const float* xr_ = x + (size_t)b * 2 * NN;
  const float* xi_ = xr_ + NN;
  for (int j = t; j < NN; j += 256) {
    const float a_ = sA[j], u = sU[j];
    const float f_ = (j == 0) ? 0.f : u - sU[j - 1];
    s_amp += a_; s_uph += u; s_frq += f_;
    s_a2 += a_ * a_; amx = fmaxf(amx, a_);
    const float xr = xr_[j], xi = xi_[j];
    const float m20r = xr*xr - xi*xi, m20i = xr*xi + xi*xr;
    const float m21r = xr*xr + xi*xi, m21i = -xr*xi + xi*xr;
    const float m22r = xr*xr - xi*xi, m22i = -xr*xi - xi*xr;
    const float m40r = m20r*m20r - m20i*m20i, m40i = m20r*m20i + m20i*m20r;
    const float m41r = m20r*m21r - m20i*m21i, m41i = m20r*m21i + m20i*m21r;
    const float m42r = m20r*m22r - m20i*m22i, m42i = m20r*m22i + m20i*m22r;
    const float m43r = m21r*m22r - m21i*m22i, m43i = m21r*m22i + m21i*m22r;
    const float m60r = m40r*m20r - m40i*m20i, m60i = m40r*m20i + m40i*m20r;
    const float m61r = m40r*m21r - m40i*m21i, m61i = m40r*m21i + m40i*m21r;
    const float m62r = m40r*m22r - m40i*m22i, m62i = m40r*m22i + m40i*m22r;
    const float m63r = m41r*m22r - m41i*m22i, m63i = m41r*m22i + m41i*m22r;
    M[0]+=m20r; M[1]+=m20i; M[2]+=m21r; M[3]+=m21i; M[4]+=m22r; M[5]+=m22i;
    M[6]+=m40r; M[7]+=m40i; M[8]+=m41r; M[9]+=m41i; M[10]+=m42r; M[11]+=m42i;
    M[12]+=m43r; M[13]+=m43i; M[14]+=m60r; M[15]+=m60i; M[16]+=m61r; M[17]+=m61i;
    M[18]+=m62r; M[19]+=m62i; M[20]+=m63r; M[21]+=m63i;
  }
  const float invN = 1.0f / (float)NN;
  const float meanA = bsum(s_amp, red) * invN;
  const float meanU = bsum(s_uph, red) * invN;
  const float meanF = bsum(s_frq, red) * invN;
  const float sumA2 = bsum(s_a2, red);
  const float ampMax = bmax(amx, red);
  float Ms[22];
  for (int k = 0; k < 22; ++k) Ms[k] = bsum(M[k], red) * invN;

  float c2[3] = {0,0,0}, c3[3] = {0,0,0}, c4[3] = {0,0,0};
  const float mean3[3] = { meanA, meanU, meanF };
  for (int j = t; j < NN; j += 256) {
    float v3[3];
    v3[0] = sA[j]; v3[1] = sU[j]; v3[2] = (j == 0) ? 0.f : sU[j] - sU[j - 1];
    #pragma unroll
    for (int ch = 0; ch < 3; ++ch) {
      const float dc = v3[ch] - mean3[ch];
      const float p2 = dc * dc, p3 = p2 * dc, p4 = p3 * dc;
      c2[ch] += p2; c3[ch] += p3; c4[ch] += p4;
    }
  }
  float cm2[3], cm3[3], cm4[3];
  for (int ch = 0; ch < 3; ++ch) {
    cm2[ch] = bsum(c2[ch], red) * invN;
    cm3[ch] = bsum(c3[ch], red) * invN;
    cm4[ch] = bsum(c4[ch], red) * invN;
  }

  if (t == 0) {
    const c2f M20 = {Ms[0], Ms[1]},  M21 = {Ms[2], Ms[3]},  M22 = {Ms[4], Ms[5]};
    const c2f M40 = {Ms[6], Ms[7]},  M41 = {Ms[8], Ms[9]},  M42 = {Ms[10], Ms[11]};
    const c2f M43 = {Ms[12], Ms[13]}, M60 = {Ms[14], Ms[15]}, M61 = {Ms[16], Ms[17]};
    const c2f M62 = {Ms[18], Ms[19]}, M63 = {Ms[20], Ms[21]};
    const c2f M20sq = cmul(M20, M20);
    const c2f M21sq = cmul(M21, M21);
    const c2f C40 = cadd(M40, cax(-3.f, M20sq));
    const c2f C41 = cadd(M41, cax(-3.f, cmul(M20, M21)));
    const float cab = M20.r * M20.r + M20.i * M20.i;
    c2f C42 = cadd(M42, cax(-2.f, M21sq)); C42.r -= cab * cab;
    c2f C60 = cadd(cadd(M60, cax(-15.f, cmul(M20, M40))), cax(30.f, cmul(M20, M20sq)));
    c2f C61 = cadd(cadd(cadd(M61, cax(-5.f, cmul(M21, M40))), cax(-10.f, cmul(M20, M41))),
                   cax(30.f, cmul(M21, M20sq)));
    c2f C62 = cadd(cadd(cadd(cadd(cadd(M62, cax(-6.f, cmul(M20, M42))),
                   cax(-8.f, cmul(M21, M41))), cax(-1.f, cmul(M22, M40))),
                   cax(6.f, cmul(M22, M20sq))), cax(24.f, cmul(M20, M21sq)));
    c2f C63 = cadd(cadd(cadd(cadd(cadd(M63, cax(-9.f, cmul(M21, M42))),
                   cax(12.f, cmul(M21, M21sq))), cax(-3.f, cmul(M20, M43))),
                   cax(-3.f, cmul(M22, M41))), cax(18.f, cmul(M20, cmul(M21, M22))));
    smom[0] = M20.r;  smom[1] = M20.i;  smom[2] = M21.r;  smom[3] = M21.i;
    smom[4] = C40.r;  smom[5] = C40.i;  smom[6] = C41.r;  smom[7] = C41.i;
    smom[8] = C42.r;  smom[9] = C42.i;  smom[10] = C60.r; smom[11] = C60.i;
    smom[12] = C61.r; smom[13] = C61.i; smom[14] = C62.r; smom[15] = C62.i;
    smom[16] = C63.r; smom[17] = C63.i;
    smom[18] = meanA; smom[19] = meanU; smom[20] = meanF;
    smom[21] = cm2[0]; smom[22] = cm2[1]; smom[23] = cm2[2];
    smom[24] = cm3[0]; smom[25] = cm3[1]; smom[26] = cm3[2];
    smom[27] = cm4[0]; smom[28] = cm4[1]; smom[29] = cm4[2];
    smom[30] = ampMax; smom[31] = sqrtf(sumA2 * invN);
  }
  __syncthreads();
  if (t < 32) {
    float acc = 0.f;
    for (int j = 0; j < 32; ++j) {
      const float mv = fminf(fmaxf(smom[j], -1.f), 1.f);
      acc += mv * Wm[t * 32 + j];
    }
    zbuf[(size_t)b * 160 + 128 + t] = fmaxf(gm[t] * acc + bm[t], 0.f);
  }
}

// ---------------------------------------------------------------------------
// Stage 4: full conv stack in LDS, each layer = implicit GEMM via f32 WMMA.
// 1 block (4 waves) per batch. Dynamic LDS: bufA(16384)+bufB(16384)+w(12288).
// ---------------------------------------------------------------------------
struct ConvPtrs { const float* w[7]; const float* g[7]; const float* b[7]; };

__global__ void __launch_bounds__(128) k_conv(
    const float* __restrict__ x, ConvPtrs cp, float* __restrict__ fbuf) {
  extern __shared__ float smem[];
  float* bufA = smem;
  float* bufB = smem + 16384;
  float* wbuf = smem + 32768;
  const int b = blockIdx.x, t = threadIdx.x;
  const float* xb = x + (size_t)b * 2 * NN;
  for (int j = t; j < 1024; j += 128) {
    const int ci = j >> 9, p = j & 511;
    float v = 0.5f * (xb[ci * NN + 2 * p] + xb[ci * NN + 2 * p + 1]);
    bufA[ci * 512 + p] = fminf(fmaxf(v, -2.f), 2.f);
  }
  float* actIn = bufA;
  float* actOut = bufB;
  const int lane = t & 31, wave = t >> 5;
  const int row = lane & 15, kh = lane >> 4, col = row;
  const int co0 = wave * 16;
  for (int l = 0; l < 7; ++l) {
    const int Cin = (l == 0) ? 2 : 64;
    const int Lin = 512 >> l, Lout = 256 >> l;
    const int Kr = Cin * 3;
    __syncthreads();
    for (int j = t; j < 64 * Kr; j += 128) wbuf[j] = cp.w[l][j];
    __syncthreads();
    const int nts = (Lout + 15) >> 4;
    const int ksteps = (Kr + 3) >> 2;
    for (int nt = 0; nt < nts; ++nt) {
      const int p0 = nt * 16;
      const int pcol = p0 + col;
      v8f acc = {};
      for (int ks = 0; ks < ksteps; ++ks) {
        const int i0 = ks * 4 + 2 * kh, i1 = i0 + 1;
        v2f a, bf;
        a.x = (i0 < Kr) ? wbuf[(co0 + row) * Kr + i0] : 0.f;
        a.y = (i1 < Kr) ? wbuf[(co0 + row) * Kr + i1] : 0.f;
        float bv0 = 0.f, bv1 = 0.f;
        if (i0 < Kr) {
          const int ci = i0 / 3, k3 = i0 - ci * 3;
          const int pos = 2 * pcol - 1 + k3;
          bv0 = (pos >= 0 && pos < Lin && pcol < Lout) ? actIn[ci * Lin + pos] : 0.f;
        }
        if (i1 < Kr) {
          const int ci = i1 / 3, k3 = i1 - ci * 3;
          const int pos = 2 * pcol - 1 + k3;
          bv1 = (pos >= 0 && pos < Lin && pcol < Lout) ? actIn[ci * Lin + pos] : 0.f;
        }
        bf.x = bv0; bf.y = bv1;
        acc = wmma_f32(a, bf, acc);
      }
      const float* gl = cp.g[l];
      const float* bl = cp.b[l];
      for (int r = 0; r < 8; ++r) {
        const int co = co0 + r + 8 * kh;
        if (pcol < Lout)
          actOut[co * Lout + pcol] = fmaxf(gl[co] * acc[r] + bl[co], 0.f);
      }
    }
    __syncthreads();
    float* tmp = actIn; actIn = actOut; actOut = tmp;
  }
  for (int j = t; j < 256; j += 128) fbuf[(size_t)b * 256 + j] = actIn[j];
}

// ---------------------------------------------------------------------------
// Generic dense layer: out[b,o] = epilogue( sum_i in[b,i] * W[o,i] )
// ---------------------------------------------------------------------------
__global__ void __launch_bounds__(256) k_dense(
    const float* __restrict__ in, int strideIn, int K,
    const float* __restrict__ W, const float* __restrict__ g,
    const float* __restrict__ bb, int Nout, int strideOut,
    float* __restrict__ out, int relu_bn) {
  const int wave = threadIdx.x >> 5, lane = threadIdx.x & 31;
  const int ntiles = (Nout + 15) >> 4;
  const int tile = blockIdx.x * 8 + wave;
  if (tile >= 64 * ntiles) return; // uniform per-wave exit
  const int b0 = (tile / ntiles) * 16, o0 = (tile % ntiles) * 16;
  const int row = lane & 15, kh = lane >> 4, col = row;
  const float* arow = in + (size_t)(b0 + row) * strideIn;
  const int oc = o0 + col;
  const int ocl = (oc < Nout) ? oc : (Nout - 1);
  const float* brow = W + (size_t)ocl * K;
  v8f acc = {};
  for (int k = 0; k < K; k += 4) {
    v2f a = *(const v2f*)(arow + k + 2 * kh);
    v2f bv = *(const v2f*)(brow + k + 2 * kh);
    acc = wmma_f32(a, bv, acc);
  }
  const float gv = g[ocl];
  const float bv = bb[ocl];
  for (int r = 0; r < 8; ++r) {
    const int bo = b0 + r + 8 * kh;
    float v = acc[r];
    v = relu_bn ? fmaxf(gv * v + bv, 0.f) : (v + bv);
    if (oc < Nout) out[(size_t)bo * strideOut + oc] = v;
  }
}

// ---------------------------------------------------------------------------
// Host launch. Input order = setup_inputs() insertion order (nested dict/list
// flattened in order): x, W_mom,g_mom,b_mom, conv_ws[0..6], conv_g[0..6],
// conv_b[0..6], W_dense,g_dense,b_dense, W_c1,g_c1,b_c1, W_c2,b_c2,
// cosm, sinm, h.
// ---------------------------------------------------------------------------
extern "C" void kernel_launch(void* const* d_in, const int* in_sizes, int n_in,
                              void* d_out, int out_size, void* d_ws, size_t ws_size,
                              hipStream_t stream) {
  (void)in_sizes; (void)n_in; (void)out_size; (void)ws_size;
  const float* x    = (const float*)d_in[0];
  const float* Wmom = (const float*)d_in[1];
  const float* gmom = (const float*)d_in[2];
  const float* bmom = (const float*)d_in[3];
  ConvPtrs cp;
  for (int i = 0; i < 7; ++i) {
    cp.w[i] = (const float*)d_in[4 + i];
    cp.g[i] = (const float*)d_in[11 + i];
    cp.b[i] = (const float*)d_in[18 + i];
  }
  const float* Wd  = (const float*)d_in[25];
  const float* gd  = (const float*)d_in[26];
  const float* bd  = (const float*)d_in[27];
  const float* Wc1 = (const float*)d_in[28];
  const float* gc1 = (const float*)d_in[29];
  const float* bc1 = (const float*)d_in[30];
  const float* Wc2 = (const float*)d_in[31];
  const float* bc2 = (const float*)d_in[32];
  const float* cm  = (const float*)d_in[33];
  const float* sm  = (const float*)d_in[34];
  const float* h   = (const float*)d_in[35];

  float* ws   = (float*)d_ws;
  float* Xre  = ws;
  float* Xim  = Xre + (size_t)BB * NN;
  float* are  = Xim + (size_t)BB * NN;
  float* aim  = are + (size_t)BB * NN;
  float* zbuf = aim + (size_t)BB * NN;        // B x 160 (f | m)
  float* ybuf = zbuf + (size_t)BB * 160;      // B x 128
  float* fbuf = ybuf + (size_t)BB * 128;      // B x 256

  k_dft_fwd<<<256, 256, 0, stream>>>(x, cm, sm, h, Xre, Xim);
  k_dft_inv<<<256, 256, 0, stream>>>(Xre, Xim, cm, sm, are, aim);
  k_moments<<<BB, 256, 0, stream>>>(x, are, aim, Wmom, gmom, bmom, zbuf);
  k_conv<<<BB, 128, 180224, stream>>>(x, cp, fbuf);
  k_dense<<<64, 256, 0, stream>>>(fbuf, 256, 256, Wd, gd, bd, 128, 160, zbuf, 1);
  k_dense<<<64, 256, 0, stream>>>(zbuf, 160, 160, Wc1, gc1, bc1, 128, 128, ybuf, 1);
  k_dense<<<16, 256, 0, stream>>>(ybuf, 128, 128, Wc2, bc2 /*unused*/, bc2, 24, 24,
                                  (float*)d_out, 0);
}